// GraphMultiHeadAttention_56538949484715
// MI455X (gfx1250) — compile-verified
//
#include <hip/hip_runtime.h>
#include <hip/hip_bf16.h>

typedef __attribute__((ext_vector_type(2))) float v2f;
typedef __attribute__((ext_vector_type(8))) float v8f;

constexpr int Bdim = 16, Ndim = 256, Ddim = 64, Edim = 64, Hdim = 8, HD = 512;

__device__ __forceinline__ v8f wmma4(v2f a, v2f b, v8f c) {
  // D = A(16x4,f32) * B(4x16,f32) + C(16x16,f32)
  return __builtin_amdgcn_wmma_f32_16x16x4_f32(false, a, false, b, (short)0, c,
                                               false, false);
}

// ---------------- K1: Q/K/V projection: [4096,64] @ [64,512] (+bias), x3 ----
__global__ __launch_bounds__(256) void qkv_kernel(
    const float* __restrict__ query,
    const float* __restrict__ Wq, const float* __restrict__ bq,
    const float* __restrict__ Wk, const float* __restrict__ bk,
    const float* __restrict__ Wv, const float* __restrict__ bv,
    float* __restrict__ Qo, float* __restrict__ Ko, float* __restrict__ Vo) {
  int wid  = blockIdx.x * 8 + (threadIdx.x >> 5);
  int lane = threadIdx.x & 31, lm = lane & 15, kh = lane >> 4;
  int w = wid >> 13;            // which weight (0..2), 8192 tiles each
  int rem = wid & 8191;
  int mtile = rem >> 5;         // 16-row tile over B*N = 4096 rows
  int ct = rem & 31;            // 16-col tile over 512 cols
  const float* W    = (w == 0) ? Wq : (w == 1) ? Wk : Wv;
  const float* bias = (w == 0) ? bq : (w == 1) ? bk : bv;
  float* Out        = (w == 0) ? Qo : (w == 1) ? Ko : Vo;
  const float* A = query + (size_t)mtile * 16 * Ddim;
  int col = ct * 16 + lm;
  v8f acc = {};
  for (int k = 0; k < Ddim; k += 4) {
    v2f a, bb;
    a.x  = A[lm * Ddim + k + 2 * kh];
    a.y  = A[lm * Ddim + k + 2 * kh + 1];
    bb.x = W[(k + 2 * kh) * HD + col];
    bb.y = W[(k + 2 * kh + 1) * HD + col];
    acc = wmma4(a, bb, acc);
  }
  float bval = bias[col];
  int h = col >> 6, d = col & 63;
#pragma unroll
  for (int g = 0; g < 8; ++g) {
    int grow = mtile * 16 + g + 8 * kh;     // global row in [0,4096)
    int b = grow >> 8, n = grow & 255;
    Out[((b * Hdim + h) * Ndim + n) * Ddim + d] = acc[g] + bval;
  }
}

// ---------------- K2: scores = Q K^T / 8 per (b,h) --------------------------
__global__ __launch_bounds__(256) void scores_kernel(
    const float* __restrict__ Q, const float* __restrict__ K,
    float* __restrict__ S) {
  int wid  = blockIdx.x * 8 + (threadIdx.x >> 5);
  int lane = threadIdx.x & 31, lm = lane & 15, kh = lane >> 4;
  int bh = wid >> 8, nt = (wid >> 4) & 15, mt = wid & 15;
  const float* Aq = Q + ((size_t)bh * Ndim + nt * 16) * Ddim;
  const float* Kb = K + ((size_t)bh * Ndim + mt * 16) * Ddim;
  v8f acc = {};
  for (int k = 0; k < Ddim; k += 4) {
    v2f a, bb;
    a.x  = Aq[lm * Ddim + k + 2 * kh];
    a.y  = Aq[lm * Ddim + k + 2 * kh + 1];
    bb.x = Kb[lm * Ddim + k + 2 * kh];      // B[k][j] = K[m0+j][k] (transposed)
    bb.y = Kb[lm * Ddim + k + 2 * kh + 1];
    acc = wmma4(a, bb, acc);
  }
#pragma unroll
  for (int g = 0; g < 8; ++g)
    S[((size_t)bh * Ndim + nt * 16 + g + 8 * kh) * Ndim + mt * 16 + lm] =
        acc[g] * 0.125f;
}

// ---------------- K3: edges @ [Wa|Wg]: bias += into scores, gate -> sigmoid -
__global__ __launch_bounds__(256) void edges_kernel(
    const float* __restrict__ edges,
    const float* __restrict__ Wa, const float* __restrict__ ba,
    const float* __restrict__ Wg, const float* __restrict__ bg,
    float* __restrict__ S, float* __restrict__ G) {
  int wid  = blockIdx.x * 8 + (threadIdx.x >> 5);
  int lane = threadIdx.x & 31, lm = lane & 15, kh = lane >> 4;
  int p0 = wid * 16;                        // 16 flattened (b,n,m) positions
  const float* A = edges + (size_t)p0 * Edim;
  v8f acc = {};
  for (int k = 0; k < Edim; k += 4) {
    v2f a, bb;
    a.x = A[lm * Edim + k + 2 * kh];
    a.y = A[lm * Edim + k + 2 * kh + 1];
    int r0 = k + 2 * kh;
    bb.x = (lm < 8) ? Wa[r0 * Hdim + lm] : Wg[r0 * Hdim + (lm - 8)];
    bb.y = (lm < 8) ? Wa[(r0 + 1) * Hdim + lm] : Wg[(r0 + 1) * Hdim + (lm - 8)];
    acc = wmma4(a, bb, acc);
  }
#pragma unroll
  for (int g = 0; g < 8; ++g) {
    int p = p0 + g + 8 * kh;
    int b = p >> 16, n = (p >> 8) & 255, m = p & 255;
    if (lm < 8) {
      size_t idx = ((size_t)(b * Hdim + lm) * Ndim + n) * Ndim + m;
      S[idx] += acc[g] + ba[lm];
    } else {
      float x = acc[g] + bg[lm - 8];
      G[((size_t)(b * Hdim + (lm - 8)) * Ndim + n) * Ndim + m] =
          1.0f / (1.0f + __expf(-x));
    }
  }
}

// ---------------- K5: edges_output[b,m,n,:] = scores[b,:,n,m] @ We + be -----
__global__ __launch_bounds__(256) void edges_out_kernel(
    const float* __restrict__ S, const float* __restrict__ We,
    const float* __restrict__ be, float* __restrict__ EO) {
  int wid  = blockIdx.x * 8 + (threadIdx.x >> 5);
  int lane = threadIdx.x & 31, lm = lane & 15, kh = lane >> 4;
  int p0 = wid * 16;                        // positions p = (b*N+m)*N + n
  int p = p0 + lm;
  int b = p >> 16, m = (p >> 8) & 255, n = p & 255;
  const float* Sp = S + ((size_t)(b * Hdim) * Ndim + n) * Ndim + m;
  const int hs = Ndim * Ndim;               // h stride in scores
  v2f a0, a1;                               // K=8 -> two k-steps
  a0.x = Sp[(0 + 2 * kh) * hs];
  a0.y = Sp[(1 + 2 * kh) * hs];
  a1.x = Sp[(4 + 2 * kh) * hs];
  a1.y = Sp[(5 + 2 * kh) * hs];
#pragma unroll
  for (int ct = 0; ct < 4; ++ct) {
    v8f acc = {};
    v2f bb;
    bb.x = We[(0 + 2 * kh) * Edim + ct * 16 + lm];
    bb.y = We[(1 + 2 * kh) * Edim + ct * 16 + lm];
    acc = wmma4(a0, bb, acc);
    bb.x = We[(4 + 2 * kh) * Edim + ct * 16 + lm];
    bb.y = We[(5 + 2 * kh) * Edim + ct * 16 + lm];
    acc = wmma4(a1, bb, acc);
    float bev = be[ct * 16 + lm];
#pragma unroll
    for (int g = 0; g < 8; ++g)
      EO[(size_t)(p0 + g + 8 * kh) * Edim + ct * 16 + lm] = acc[g] + bev;
  }
}

// ---------------- K4: row softmax over scores (in place), wave per row ------
__global__ __launch_bounds__(256) void softmax_kernel(float* __restrict__ S) {
  int row  = blockIdx.x * 8 + (threadIdx.x >> 5);
  int lane = threadIdx.x & 31;
  float* p = S + (size_t)row * Ndim;
  float v[8];
  float mx = -3.4e38f;
#pragma unroll
  for (int i = 0; i < 8; ++i) { v[i] = p[lane + 32 * i]; mx = fmaxf(mx, v[i]); }
#pragma unroll
  for (int o = 16; o > 0; o >>= 1) mx = fmaxf(mx, __shfl_xor(mx, o, 32));
  float s = 0.f;
#pragma unroll
  for (int i = 0; i < 8; ++i) { v[i] = __expf(v[i] - mx); s += v[i]; }
#pragma unroll
  for (int o = 16; o > 0; o >>= 1) s += __shfl_xor(s, o, 32);
  float inv = 1.0f / s;
#pragma unroll
  for (int i = 0; i < 8; ++i) p[lane + 32 * i] = v[i] * inv;
}

// ---------------- K6: Vg = g @ V per (b,h): 256x64, K=256 -------------------
__global__ __launch_bounds__(256) void vg_kernel(
    const float* __restrict__ G, const float* __restrict__ V,
    float* __restrict__ Vg) {
  int wid  = blockIdx.x * 8 + (threadIdx.x >> 5);
  int lane = threadIdx.x & 31, lm = lane & 15, kh = lane >> 4;
  int bh = wid >> 6, nt = (wid >> 2) & 15, dt = wid & 3;
  const float* A  = G + ((size_t)bh * Ndim + nt * 16) * Ndim;
  const float* Bp = V + (size_t)bh * Ndim * Ddim + dt * 16;
  v8f acc = {};
  for (int k = 0; k < Ndim; k += 4) {
    v2f a, bb;
    a.x  = A[lm * Ndim + k + 2 * kh];
    a.y  = A[lm * Ndim + k + 2 * kh + 1];
    bb.x = Bp[(k + 2 * kh) * Ddim + lm];
    bb.y = Bp[(k + 2 * kh + 1) * Ddim + lm];
    acc = wmma4(a, bb, acc);
  }
#pragma unroll
  for (int g = 0; g < 8; ++g)
    Vg[((size_t)bh * Ndim + nt * 16 + g + 8 * kh) * Ddim + dt * 16 + lm] = acc[g];
}

// ---------------- K7: O = attn @ Vg per (b,h), packed [B,N,H*D] -------------
__global__ __launch_bounds__(256) void av_kernel(
    const float* __restrict__ Attn, const float* __restrict__ Vg,
    float* __restrict__ O) {
  int wid  = blockIdx.x * 8 + (threadIdx.x >> 5);
  int lane = threadIdx.x & 31, lm = lane & 15, kh = lane >> 4;
  int bh = wid >> 6, nt = (wid >> 2) & 15, dt = wid & 3;
  const float* A  = Attn + ((size_t)bh * Ndim + nt * 16) * Ndim;
  const float* Bp = Vg + (size_t)bh * Ndim * Ddim + dt * 16;
  v8f acc = {};
  for (int k = 0; k < Ndim; k += 4) {
    v2f a, bb;
    a.x  = A[lm * Ndim + k + 2 * kh];
    a.y  = A[lm * Ndim + k + 2 * kh + 1];
    bb.x = Bp[(k + 2 * kh) * Ddim + lm];
    bb.y = Bp[(k + 2 * kh + 1) * Ddim + lm];
    acc = wmma4(a, bb, acc);
  }
  int b = bh >> 3, h = bh & 7;
#pragma unroll
  for (int g = 0; g < 8; ++g) {
    int n = nt * 16 + g + 8 * kh;
    O[((size_t)(b * Ndim + n)) * HD + h * Ddim + dt * 16 + lm] = acc[g];
  }
}

// ---------------- K8: attention_output = O @ Wo + bo (4096x512x64) ----------
__global__ __launch_bounds__(256) void outproj_kernel(
    const float* __restrict__ O, const float* __restrict__ Wo,
    const float* __restrict__ bo, float* __restrict__ AO) {
  int wid  = blockIdx.x * 8 + (threadIdx.x >> 5);
  int lane = threadIdx.x & 31, lm = lane & 15, kh = lane >> 4;
  int mt = wid >> 2, ct = wid & 3;
  const float* A = O + (size_t)mt * 16 * HD;
  v8f acc = {};
  for (int k = 0; k < HD; k += 4) {
    v2f a, bb;
    a.x  = A[lm * HD + k + 2 * kh];
    a.y  = A[lm * HD + k + 2 * kh + 1];
    bb.x = Wo[(k + 2 * kh) * Ddim + ct * 16 + lm];
    bb.y = Wo[(k + 2 * kh + 1) * Ddim + ct * 16 + lm];
    acc = wmma4(a, bb, acc);
  }
  float bov = bo[ct * 16 + lm];
#pragma unroll
  for (int g = 0; g < 8; ++g)
    AO[(size_t)(mt * 16 + g + 8 * kh) * Ddim + ct * 16 + lm] = acc[g] + bov;
}

extern "C" void kernel_launch(void* const* d_in, const int* in_sizes, int n_in,
                              void* d_out, int out_size, void* d_ws,
                              size_t ws_size, hipStream_t stream) {
  const float* query = (const float*)d_in[0];
  const float* edges = (const float*)d_in[1];
  const float* Wq = (const float*)d_in[2];  const float* bq = (const float*)d_in[3];
  const float* Wk = (const float*)d_in[4];  const float* bk = (const float*)d_in[5];
  const float* Wv = (const float*)d_in[6];  const float* bv = (const float*)d_in[7];
  const float* Wa = (const float*)d_in[8];  const float* ba = (const float*)d_in[9];
  const float* Wg = (const float*)d_in[10]; const float* bg = (const float*)d_in[11];
  const float* Wo = (const float*)d_in[12]; const float* bo = (const float*)d_in[13];
  const float* We = (const float*)d_in[14]; const float* be = (const float*)d_in[15];

  float* ws = (float*)d_ws;
  const size_t QS = (size_t)Bdim * Hdim * Ndim * Ddim;   // 2,097,152 floats
  const size_t SS = (size_t)Bdim * Hdim * Ndim * Ndim;   // 8,388,608 floats
  float* Q  = ws;
  float* K  = Q + QS;
  float* V  = K + QS;
  float* S  = V + QS;          // scores [B,H,N,N]; later reused as attn
  float* G  = S + SS;          // sigmoid gate [B,H,N,N]
  float* Vg = G + SS;          // [B,H,N,D]
  float* O  = Vg + QS;         // [B,N,H*D]

  float* AO = (float*)d_out;                             // [B,N,D]
  float* EO = AO + (size_t)Bdim * Ndim * Ddim;           // [B,N,N,E]

  qkv_kernel<<<3072, 256, 0, stream>>>(query, Wq, bq, Wk, bk, Wv, bv, Q, K, V);
  scores_kernel<<<4096, 256, 0, stream>>>(Q, K, S);
  edges_kernel<<<8192, 256, 0, stream>>>(edges, Wa, ba, Wg, bg, S, G);
  edges_out_kernel<<<8192, 256, 0, stream>>>(S, We, be, EO);  // before softmax
  softmax_kernel<<<4096, 256, 0, stream>>>(S);                // in-place -> attn
  vg_kernel<<<1024, 256, 0, stream>>>(G, V, Vg);
  av_kernel<<<1024, 256, 0, stream>>>(S, Vg, O);
  outproj_kernel<<<128, 256, 0, stream>>>(O, Wo, bo, AO);
}